// MaskRCNNMaskProposal_17892833755385
// MI455X (gfx1250) — compile-verified
//
#include <hip/hip_runtime.h>

typedef __attribute__((ext_vector_type(2))) float v2f;
typedef __attribute__((ext_vector_type(4))) float v4f;
typedef __attribute__((ext_vector_type(8))) float v8f;

#define POOL 14
#define XW   162           // Wx row stride (even => b64-aligned A reads; bank-friendly vs 64 banks)
#define NCH  256

__global__ __launch_bounds__(32)
void roi_align_fpn_wmma(const float* __restrict__ f0, const float* __restrict__ f1,
                        const float* __restrict__ f2, const float* __restrict__ f3,
                        const float* __restrict__ boxes, float* __restrict__ out,
                        int nper /* rois per batch image */)
{
    __shared__ __align__(16) float Wx[16 * XW];

    const int lane = threadIdx.x;        // 0..31
    const int cblk = blockIdx.x;         // 0..15 (16-channel blocks)
    const int roi  = blockIdx.y;         // 0..255
    const int bat  = roi / nper;

    // ---- box + FPN level mapper (uniform; boxes addr is uniform -> SMEM load) ----
    const float* bx = boxes + (size_t)roi * 4;
    const float bx1 = bx[0], by1 = bx[1], bx2 = bx[2], by2 = bx[3];
    const float s = sqrtf((bx2 - bx1) * (by2 - by1));
    const float lvlf = floorf(4.0f + log2f(s * (1.0f / 224.0f) + 1e-6f));
    int lvl = (int)lvlf;
    lvl = lvl < 2 ? 2 : (lvl > 5 ? 5 : lvl);
    const int idx = __builtin_amdgcn_readfirstlane(lvl - 2);      // 0..3, force SGPR
    const float scale = 1.0f / (float)(4 << idx);                 // 0.25 .. 0.03125
    const int   L = 200 >> idx;                                   // H == W: 200,100,50,25
    const float Lf = (float)L;
    const float* fI = (idx == 0) ? f0 : (idx == 1) ? f1 : (idx == 2) ? f2 : f3;
    const float* fbase = fI + (size_t)bat * NCH * L * L;

    const float x1s = bx1 * scale, y1s = by1 * scale;
    const float rw = fmaxf(bx2 * scale - x1s, 1.0f);
    const float rh = fmaxf(by2 * scale - y1s, 1.0f);
    const float bw = rw * (1.0f / POOL);
    const float bh = rh * (1.0f / POOL);

    // ---- x-region bounds from first/last sample (xs monotonic, bw > 0) ----
    float c0 = fminf(fmaxf(x1s + bw * 0.25f, 0.0f), Lf - 1.0f);
    int xlo0 = (int)floorf(c0); xlo0 = xlo0 > L - 2 ? L - 2 : (xlo0 < 0 ? 0 : xlo0);
    float cN = fminf(fmaxf(x1s + bw * 13.875f, 0.0f), Lf - 1.0f);
    int xloN = (int)floorf(cN); xloN = xloN > L - 2 ? L - 2 : (xloN < 0 ? 0 : xloN);
    const int x0e = __builtin_amdgcn_readfirstlane(xlo0 & ~1);    // even start: aligned b64
    const int xspan = (xloN + 1) - x0e + 1;                       // <= 154 < XW
    const int nchunk = __builtin_amdgcn_readfirstlane((xspan + 3) >> 2);
    int nf = ((L - 4 - x0e) >> 2) + 1;                            // chunks fully inside [0, L-1]
    nf = nf < 0 ? 0 : (nf > nchunk ? nchunk : nf);
    const int nfull = __builtin_amdgcn_readfirstlane(nf);

    // ---- build A = Wx[16 pw x xspan] in LDS (b128 zero + ds_add_f32 scatter) ----
    {
        v4f* wz = (v4f*)Wx;
        for (int i = lane; i < (16 * XW) / 4; i += 32) wz[i] = (v4f){0.f, 0.f, 0.f, 0.f};
    }
    __syncthreads();
    if (lane < 2 * POOL) {
        const float off = ((float)lane + 0.5f) * 0.5f;
        const float cx  = x1s + bw * off;
        const bool  vx  = (cx > -1.0f) && (cx < Lf);
        float cc = fminf(fmaxf(cx, 0.0f), Lf - 1.0f);
        int lo = (int)floorf(cc); lo = lo > L - 2 ? L - 2 : (lo < 0 ? 0 : lo);
        const float fr = cc - (float)lo;
        if (vx) {
            const int row = lane >> 1;               // pw this sample belongs to
            atomicAdd(&Wx[row * XW + (lo - x0e)],     0.5f * (1.0f - fr));
            atomicAdd(&Wx[row * XW + (lo - x0e) + 1], 0.5f * fr);
        }
    }
    __syncthreads();

    // ---- per-lane WMMA operand roles ----
    const int cofs = (lane & 15);                    // channel within block / B column
    const int koff = (lane >> 4) << 1;               // K-pair: lanes 0-15 -> K0,1; 16-31 -> K2,3
    const int ch   = cblk * 16 + cofs;
    const int cbe  = ch * L * L;                     // per-lane element offset (fits i32)
    float* op = out + ((size_t)roi * NCH + ch) * (POOL * POOL);
    const float* wrow = &Wx[cofs * XW];

    for (int ph = 0; ph < POOL; ++ph) {
        // y taps of this bin's two sample rows (uniform)
        float cya = y1s + bh * ((float)ph + 0.25f);
        float cyb = y1s + bh * ((float)ph + 0.75f);
        const bool va = (cya > -1.0f) && (cya < Lf);
        const bool vb = (cyb > -1.0f) && (cyb < Lf);
        cya = fminf(fmaxf(cya, 0.0f), Lf - 1.0f);
        cyb = fminf(fmaxf(cyb, 0.0f), Lf - 1.0f);
        int ya = (int)floorf(cya); ya = ya > L - 2 ? L - 2 : (ya < 0 ? 0 : ya);
        int yb = (int)floorf(cyb); yb = yb > L - 2 ? L - 2 : (yb < 0 ? 0 : yb);
        ya = __builtin_amdgcn_readfirstlane(ya);
        yb = __builtin_amdgcn_readfirstlane(yb);
        const float fa = cya - (float)ya;
        const float fb = cyb - (float)yb;
        const float w00 = va ? 0.5f * (1.0f - fa) : 0.0f;
        const float w01 = va ? 0.5f * fa          : 0.0f;
        const float w10 = vb ? 0.5f * (1.0f - fb) : 0.0f;
        const float w11 = vb ? 0.5f * fb          : 0.0f;
        // uniform row bases (saddr) + per-lane 32-bit element offset (vaddr)
        const float* r0 = fbase + (size_t)ya * L;
        const float* r1 = r0 + L;
        const float* r2 = fbase + (size_t)yb * L;
        const float* r3 = r2 + L;

        v8f acc = {};
        int eoff = cbe + x0e + koff;
        // main loop: all 4 columns of each chunk are in-bounds
        for (int kx = 0; kx < nfull; ++kx, eoff += 4) {
            const v2f p0 = *(const v2f*)(r0 + eoff);
            const v2f p1 = *(const v2f*)(r1 + eoff);
            const v2f p2 = *(const v2f*)(r2 + eoff);
            const v2f p3 = *(const v2f*)(r3 + eoff);
            const v2f bb = w00 * p0 + w01 * p1 + w10 * p2 + w11 * p3;  // y-interp row
            const v2f av = *(const v2f*)(wrow + kx * 4 + koff);
            acc = __builtin_amdgcn_wmma_f32_16x16x4_f32(
                false, av, false, bb, (short)0, acc, false, false);
        }
        // tail: clamp pad columns to L-1 (their A-weights are zero)
        for (int kx = nfull; kx < nchunk; ++kx) {
            const int xb0 = x0e + kx * 4 + koff;
            const int xa0 = xb0     < L - 1 ? xb0     : L - 1;
            const int xa1 = xb0 + 1 < L - 1 ? xb0 + 1 : L - 1;
            const float b0 = w00 * r0[cbe + xa0] + w01 * r1[cbe + xa0]
                           + w10 * r2[cbe + xa0] + w11 * r3[cbe + xa0];
            const float b1 = w00 * r0[cbe + xa1] + w01 * r1[cbe + xa1]
                           + w10 * r2[cbe + xa1] + w11 * r3[cbe + xa1];
            v2f bb; bb.x = b0; bb.y = b1;
            const v2f av = *(const v2f*)(wrow + kx * 4 + koff);
            acc = __builtin_amdgcn_wmma_f32_16x16x4_f32(
                false, av, false, bb, (short)0, acc, false, false);
        }

        // D layout: vgpr r, lane l -> (pw = r + 8*(l>>4), c = l&15); rows 14,15 are zero
        const int half = lane >> 4;
        #pragma unroll
        for (int r = 0; r < 8; ++r) {
            const int pw = r + 8 * half;
            if (pw < POOL) op[ph * POOL + pw] = acc[r];
        }
    }
}

extern "C" void kernel_launch(void* const* d_in, const int* in_sizes, int n_in,
                              void* d_out, int out_size, void* d_ws, size_t ws_size,
                              hipStream_t stream) {
    const float* f0 = (const float*)d_in[0];
    const float* f1 = (const float*)d_in[1];
    const float* f2 = (const float*)d_in[2];
    const float* f3 = (const float*)d_in[3];
    const float* boxes = (const float*)d_in[4];
    float* out = (float*)d_out;

    const int K = in_sizes[4] / 4;   // total rois (B*N = 256)
    const int nper = 128;            // N per image (B = 2)

    dim3 grid(16 /* channel blocks of 16 */, K);
    dim3 block(32, 1, 1);            // one wave32 per WG: EXEC all-1s at WMMA
    roi_align_fpn_wmma<<<grid, block, 0, stream>>>(f0, f1, f2, f3, boxes, out, nper);
}